// AttenComm_65601330479407
// MI455X (gfx1250) — compile-verified
//
#include <hip/hip_runtime.h>
#include <hip/hip_bf16.h>

#define B_  4
#define N_  5
#define C_  64
#define H_  96
#define W_  288
#define HW_ (H_*W_)

// softmax([1.0, 0.5]) decay weights, precomputed
#define DECAY0 0.6224593312f
#define DECAY1 0.3775406688f

typedef float v2f __attribute__((ext_vector_type(2)));
typedef float v8f __attribute__((ext_vector_type(8)));

#if defined(__has_builtin)
#if __has_builtin(__builtin_amdgcn_wmma_f32_16x16x4_f32)
#define HAVE_WMMA_F32X4 1
#endif
#endif

// ---------------------------------------------------------------------------
// Kernel 1: per-pixel frame score for both history frames -> binary mask (u8).
// All 4 non-ego agents share one score; ego conf is identically 1 (>0.5).
// sigmoid(ctx @ w) folds: ctx.w = sum_k attn_k * (w . v_k)  -> single sweep.
// Fully coalesced: lane = pixel, channel/agent strides are uniform.
// ---------------------------------------------------------------------------
__device__ __forceinline__ float frame_sc(const float* __restrict__ x, int b, int p,
                                          const float* __restrict__ mlp_w, float bias)
{
    const float* base = x + (size_t)b * N_ * C_ * HW_ + p;
    float s1=0.f,s2=0.f,s3=0.f,s4=0.f;
    float d1=0.f,d2=0.f,d3=0.f,d4=0.f;
    for (int c = 0; c < C_; ++c) {
        const float wm = mlp_w[c];                 // uniform -> scalar load
        const float* bc = base + (size_t)c * HW_;
        float e  = bc[0];
        float v1 = bc[(size_t)1*C_*HW_];
        float v2 = bc[(size_t)2*C_*HW_];
        float v3 = bc[(size_t)3*C_*HW_];
        float v4 = bc[(size_t)4*C_*HW_];
        s1 = fmaf(e, v1, s1);  d1 = fmaf(wm, v1, d1);
        s2 = fmaf(e, v2, s2);  d2 = fmaf(wm, v2, d2);
        s3 = fmaf(e, v3, s3);  d3 = fmaf(wm, v3, d3);
        s4 = fmaf(e, v4, s4);  d4 = fmaf(wm, v4, d4);
    }
    s1 *= 0.125f; s2 *= 0.125f; s3 *= 0.125f; s4 *= 0.125f;  // / sqrt(C)
    float m = fmaxf(fmaxf(s1,s2), fmaxf(s3,s4));
    float e1 = expf(s1-m), e2 = expf(s2-m), e3 = expf(s3-m), e4 = expf(s4-m);
    float inv = 1.f / (e1+e2+e3+e4);
    float ctxw = (e1*d1 + e2*d2 + e3*d3 + e4*d4) * inv + bias;
    return 1.f / (1.f + expf(-ctxw));             // sigmoid
}

__global__ __launch_bounds__(256)
void score_kernel(const float* __restrict__ x0, const float* __restrict__ x1,
                  const float* __restrict__ mlp_w, const float* __restrict__ mlp_b,
                  unsigned char* __restrict__ maskpre)
{
    const int gid = blockIdx.x * blockDim.x + threadIdx.x;   // exactly B*HW threads
    const int b = gid / HW_;
    const int p = gid % HW_;
    const float bias = mlp_b[0];
    float conf = DECAY0 * frame_sc(x0, b, p, mlp_w, bias)
               + DECAY1 * frame_sc(x1, b, p, mlp_w, bias);
    maskpre[gid] = (conf > 0.5f) ? 1u : 0u;
}

// ---------------------------------------------------------------------------
// Kernel 2: 3x3 max-pool (SAME) on the binary mask == OR-dilation.
// ---------------------------------------------------------------------------
__global__ __launch_bounds__(256)
void dilate_kernel(const unsigned char* __restrict__ mp,
                   unsigned char* __restrict__ dm)
{
    const int gid = blockIdx.x * blockDim.x + threadIdx.x;
    const int b = gid / HW_;
    const int p = gid % HW_;
    const int h = p / W_, w = p % W_;
    unsigned m = 0;
    #pragma unroll
    for (int dy = -1; dy <= 1; ++dy) {
        int hh = h + dy;
        if (hh < 0 || hh >= H_) continue;
        #pragma unroll
        for (int dx = -1; dx <= 1; ++dx) {
            int ww = w + dx;
            if (ww < 0 || ww >= W_) continue;
            m |= mp[b * HW_ + hh * W_ + ww];
        }
    }
    dm[gid] = (unsigned char)m;
}

// ---------------------------------------------------------------------------
// Kernel 3: warp (masked bilinear, affine theta) + 5-agent attention fuse.
// One wave = 16 pixels. Lane L owns pixel (L&15); channels c with c%4 in
// {0,1} for L<16 and {2,3} for L>=16. This residency IS the A(16x4)/B(4x16)
// f32 WMMA operand layout (A: V0=K0|K2, V1=K1|K3; B mirrors), so the five
// 16x16 Grams D_n = E * Vn^T accumulate with v_wmma_f32_16x16x4_f32 and the
// needed dots s[p,n] are the diagonals. EXEC stays all-ones (exact grid).
//
// Addressing: b is derived from blockIdx only (scalar base in SGPRs); the
// lane-varying part (tap offset + half-channel skew) is ONE 32-bit byte
// offset precomputed per (agent, tap) -> global_load saddr+voffset form,
// delta-channel folded into the 24-bit instruction offset. No 64-bit vector
// address math, no spills of index pairs.
// ---------------------------------------------------------------------------
#define BLOCKS_PER_B (HW_ / 128)   // 216: 8 waves/block * 16 px/wave

__global__ __launch_bounds__(256)
void warp_fuse_kernel(const float* __restrict__ x0,
                      const float* __restrict__ pt,   // (B,N,N,4,4)
                      const unsigned char* __restrict__ dm,
                      float* __restrict__ out)        // (B,C,H,W)
{
    const int lane = threadIdx.x & 31;
    const int q    = lane & 15;      // pixel within 16-group
    const int half = lane >> 4;      // channel interleave half
    const int b    = blockIdx.x / BLOCKS_PER_B;                       // block-uniform
    const int grp  = (blockIdx.x % BLOCKS_PER_B) * 8 + (threadIdx.x >> 5);
    const int p = grp * 16 + q;
    const int h = p / W_;
    const int w = p % W_;

    const float gx = -1.f + 2.f * (float)w / (float)(W_ - 1);
    const float gy = -1.f + 2.f * (float)h / (float)(H_ - 1);

    // Per-agent taps: 32-bit BYTE offset (incl. half-channel skew) + weight
    int   voffb[N_][4];
    float twt [N_][4];
    const unsigned char* dmb = dm + (size_t)b * HW_;
    #pragma unroll
    for (int n = 0; n < N_; ++n) {
        const float* M = pt + (((size_t)b * N_ + 0) * N_ + n) * 16;  // theta = t[:,0]
        float t00 = M[0];
        float t01 = M[1] * ((float)H_ / (float)W_);
        float t02 = M[3] * (2.0f / (4.0f * 0.4f * (float)W_));
        float t10 = M[4] * ((float)W_ / (float)H_);
        float t11 = M[5];
        float t12 = M[7] * (2.0f / (4.0f * 0.4f * (float)H_));
        float px = (t00*gx + t01*gy + t02 + 1.f) * 0.5f * (float)(W_ - 1);
        float py = (t10*gx + t11*gy + t12 + 1.f) * 0.5f * (float)(H_ - 1);
        float x0f = floorf(px), y0f = floorf(py);
        float wx = px - x0f,    wy = py - y0f;
        #pragma unroll
        for (int t = 0; t < 4; ++t) {
            float xx = x0f + (float)(t & 1);
            float yy = y0f + (float)(t >> 1);
            float bw = ((t & 1) ? wx : 1.f - wx) * ((t >> 1) ? wy : 1.f - wy);
            float valid = (yy >= 0.f && yy <= (float)(H_-1) &&
                           xx >= 0.f && xx <= (float)(W_-1)) ? 1.f : 0.f;
            int yi = (int)fminf(fmaxf(yy, 0.f), (float)(H_ - 1));
            int xi = (int)fminf(fmaxf(xx, 0.f), (float)(W_ - 1));
            float mk = (n == 0) ? 1.f : (float)dmb[yi * W_ + xi];   // ego mask == 1
            voffb[n][t] = (yi * W_ + xi + half * (2 * HW_)) * 4;
            twt [n][t] = bw * valid * mk;
        }
    }

    // Scalar (SGPR) image base for this batch; all lane-variation is in voffb.
    const char* xw = (const char*)(x0 + (size_t)b * N_ * C_ * HW_);

    // Sample the lane's channel pair (4j+2*half, 4j+2*half+1) of agent n.
    auto sample_pair = [&](int n, int jbase, float& r0, float& r1) {
        const char* bn0 = xw + ((size_t)n * C_ + jbase) * (size_t)(HW_ * 4);
        const char* bn1 = bn0 + (size_t)(HW_ * 4);     // folds into inst offset
        r0 = fmaf(twt[n][0], *(const float*)(bn0 + voffb[n][0]),
             fmaf(twt[n][1], *(const float*)(bn0 + voffb[n][1]),
             fmaf(twt[n][2], *(const float*)(bn0 + voffb[n][2]),
                  twt[n][3] * *(const float*)(bn0 + voffb[n][3]))));
        r1 = fmaf(twt[n][0], *(const float*)(bn1 + voffb[n][0]),
             fmaf(twt[n][1], *(const float*)(bn1 + voffb[n][1]),
             fmaf(twt[n][2], *(const float*)(bn1 + voffb[n][2]),
                  twt[n][3] * *(const float*)(bn1 + voffb[n][3]))));
    };

    // ---- Pass 1: Gram accumulators D_n = E * Vn^T over 16 K-chunks of 4 ----
    float s[N_];
#if HAVE_WMMA_F32X4
    v8f D[N_];
    #pragma unroll
    for (int n = 0; n < N_; ++n) D[n] = (v8f){0.f,0.f,0.f,0.f,0.f,0.f,0.f,0.f};
    for (int j = 0; j < 16; ++j) {
        float a0, a1; sample_pair(0, 4*j, a0, a1);     // ego (agent 0)
        v2f A = {a0, a1};
        D[0] = __builtin_amdgcn_wmma_f32_16x16x4_f32(false, A, false, A,
                                                     (short)0, D[0], false, false);
        #pragma unroll
        for (int n = 1; n < N_; ++n) {
            float b0, b1; sample_pair(n, 4*j, b0, b1);
            v2f Bm = {b0, b1};
            D[n] = __builtin_amdgcn_wmma_f32_16x16x4_f32(false, A, false, Bm,
                                                         (short)0, D[n], false, false);
        }
    }
    // Diagonal extraction: diag of pixel p<8 lives at lane p vgpr p;
    // p>=8 at lane p+16 vgpr p-8. Select own vgpr r=lane&7, bpermute-broadcast.
    const int r = lane & 7;
    const int srcLane = (q < 8) ? q : (q + 16);
    #pragma unroll
    for (int n = 0; n < N_; ++n) {
        float sel = D[n][0];
        #pragma unroll
        for (int j = 1; j < 8; ++j) sel = (r == j) ? D[n][j] : sel;
        s[n] = __shfl(sel, srcLane, 32) * 0.125f;              // / sqrt(C)
    }
#else
    // Fallback: per-lane partial dots over owned channels, pair-reduce via xor-16
    float acc[N_];
    #pragma unroll
    for (int n = 0; n < N_; ++n) acc[n] = 0.f;
    for (int j = 0; j < 16; ++j) {
        float e0, e1; sample_pair(0, 4*j, e0, e1);
        acc[0] = fmaf(e0, e0, fmaf(e1, e1, acc[0]));
        #pragma unroll
        for (int n = 1; n < N_; ++n) {
            float v0, v1; sample_pair(n, 4*j, v0, v1);
            acc[n] = fmaf(e0, v0, fmaf(e1, v1, acc[n]));
        }
    }
    #pragma unroll
    for (int n = 0; n < N_; ++n)
        s[n] = (acc[n] + __shfl_xor(acc[n], 16, 32)) * 0.125f;
#endif

    // softmax over the 5 agents
    float m = s[0];
    #pragma unroll
    for (int n = 1; n < N_; ++n) m = fmaxf(m, s[n]);
    float aw[N_]; float sum = 0.f;
    #pragma unroll
    for (int n = 0; n < N_; ++n) { aw[n] = expf(s[n] - m); sum += aw[n]; }
    float inv = 1.f / sum;
    #pragma unroll
    for (int n = 0; n < N_; ++n) aw[n] *= inv;

    // ---- Pass 2: resample (L2-hot) and emit ctx, coalesced (B,C,H,W) ----
    // Store address: scalar base (b, 4j) + one precomputed lane byte offset.
    char* ob = (char*)out + (size_t)b * C_ * HW_ * 4;
    const int vso = (p + half * (2 * HW_)) * 4;
    for (int j = 0; j < 16; ++j) {
        float cx0 = 0.f, cx1 = 0.f;
        #pragma unroll
        for (int n = 0; n < N_; ++n) {
            float v0, v1; sample_pair(n, 4*j, v0, v1);
            cx0 = fmaf(aw[n], v0, cx0);
            cx1 = fmaf(aw[n], v1, cx1);
        }
        char* o0 = ob + (size_t)(4*j) * (size_t)(HW_ * 4);
        *(float*)(o0 + vso)             = cx0;
        *(float*)(o0 + (HW_ * 4) + vso) = cx1;
    }
}

// ---------------------------------------------------------------------------
extern "C" void kernel_launch(void* const* d_in, const int* in_sizes, int n_in,
                              void* d_out, int out_size, void* d_ws, size_t ws_size,
                              hipStream_t stream)
{
    const float* x0    = (const float*)d_in[0];
    const float* x1    = (const float*)d_in[1];
    const float* pt    = (const float*)d_in[2];
    const float* mlp_w = (const float*)d_in[3];
    const float* mlp_b = (const float*)d_in[4];
    // d_in[5] = record_len (always N, unused)
    float* out = (float*)d_out;

    unsigned char* maskpre = (unsigned char*)d_ws;               // B*HW bytes
    unsigned char* dilm    = maskpre + (size_t)B_ * HW_;          // B*HW bytes

    const int totalPix = B_ * HW_;                                // 110592
    score_kernel<<<totalPix / 256, 256, 0, stream>>>(x0, x1, mlp_w, mlp_b, maskpre);
    dilate_kernel<<<totalPix / 256, 256, 0, stream>>>(maskpre, dilm);
    // 1 wave = 16 pixels; 8 waves/block -> 128 px/block; exact grid, EXEC all-ones
    warp_fuse_kernel<<<totalPix / 128, 256, 0, stream>>>(x0, pt, dilm, out);
    (void)in_sizes; (void)n_in; (void)out_size; (void)ws_size;
}